// ParticleGNN_63702954934413
// MI455X (gfx1250) — compile-verified
//
#include <hip/hip_runtime.h>
#include <hip/hip_bf16.h>
#include <math.h>

typedef __attribute__((ext_vector_type(2))) float v2f;
typedef __attribute__((ext_vector_type(8))) float v8f;

#define HID 128
#define NG  64
#define SWT_LD 132   // padded LDS row stride (dwords): bank-conflict-free B loads

// ---------------------------------------------------------------------------
// utility: zero a buffer (count must be a multiple of 4 floats)
// ---------------------------------------------------------------------------
__global__ void zero_kernel(float4* __restrict__ p, int n4) {
    int i = blockIdx.x * blockDim.x + threadIdx.x;
    if (i < n4) p[i] = make_float4(0.f, 0.f, 0.f, 0.f);
}

// ---------------------------------------------------------------------------
// degree: deg[dst] += 1 for every edge (buffer pre-zeroed)
// ---------------------------------------------------------------------------
__global__ void deg_kernel(const int* __restrict__ dst, float* __restrict__ deg, int E) {
    int e = blockIdx.x * blockDim.x + threadIdx.x;
    if (e < E) atomicAdd(&deg[dst[e]], 1.0f);
}

// dinv[i] = 1/sqrt(deg[i] + 1)   (self loop)
__global__ void dinv_kernel(float* __restrict__ d, int N) {
    int i = blockIdx.x * blockDim.x + threadIdx.x;
    if (i < N) d[i] = rsqrtf(d[i] + 1.0f);
}

// ---------------------------------------------------------------------------
// GEMM  Out[N,128] = A[N,128] * W[128,128]   via V_WMMA_F32_16X16X4_F32
// one wave per 16-row M tile; W held transposed in LDS (sWt[n*SWT_LD+k]) so a
// B fragment (k,k+1 for fixed n) is one ds_load_b64; with stride 132 dwords
// lane l hits bank 4*(l%16)+khalf -> conflict-free across the wave.
// A frag layout (ISA 7.12.2, 32-bit A 16x4): lane = M%16 + 16*(K/2), vgpr=K%2.
// C/D layout: vgpr r -> row r (lanes 0-15) / row r+8 (lanes 16-31), col = lane%16.
// ---------------------------------------------------------------------------
__global__ __launch_bounds__(256) void gemm128_kernel(
    const float* __restrict__ A, const float* __restrict__ W,
    float* __restrict__ Out, int nrows)
{
    __shared__ float sWt[HID * SWT_LD];  // ~66 KB, transposed + padded
    for (int i = threadIdx.x; i < HID * HID; i += blockDim.x) {
        int k = i >> 7, n = i & 127;
        sWt[n * SWT_LD + k] = W[i];
    }
    __syncthreads();

    const int wave  = threadIdx.x >> 5;
    const int lane  = threadIdx.x & 31;
    const int mbase = (blockIdx.x * 8 + wave) * 16;
    if (mbase >= nrows) return;

    const int mrow  = mbase + (lane & 15);
    const int khalf = (lane >> 4) * 2;          // 0 or 2
    const int ncol  = lane & 15;

    // load all A fragments for K=128 (32 k-steps of 4)
    v2f a[32];
    const float* arow = A + (size_t)min(mrow, nrows - 1) * HID;
    #pragma unroll
    for (int ks = 0; ks < 32; ++ks) {
        float2 t = *(const float2*)(arow + ks * 4 + khalf);
        a[ks].x = t.x; a[ks].y = t.y;
    }

    const int rbase = mbase + ((lane >> 4) << 3);   // +0 or +8
    for (int nt = 0; nt < 8; ++nt) {
        v8f c = {};
        const float* bcol = sWt + (size_t)(nt * 16 + ncol) * SWT_LD + khalf;
        #pragma unroll
        for (int ks = 0; ks < 32; ++ks) {
            float2 t = *(const float2*)(bcol + ks * 4);
            v2f b; b.x = t.x; b.y = t.y;
            c = __builtin_amdgcn_wmma_f32_16x16x4_f32(
                    false, a[ks], false, b, (short)0, c, false, false);
        }
        #pragma unroll
        for (int r = 0; r < 8; ++r) {
            int row = rbase + r;
            if (row < nrows) Out[(size_t)row * HID + nt * 16 + ncol] = c[r];
        }
    }
}

// GEMM  Out[N,128] = X[N,4] * W[4,128]  (layer 1, K=4 -> single WMMA per tile)
// sWt stride 4: lane l reads bank 4*(l%16)+khalf -> already conflict-free.
__global__ __launch_bounds__(256) void gemm4_kernel(
    const float* __restrict__ X, const float* __restrict__ W,
    float* __restrict__ Out, int nrows)
{
    __shared__ float sWt[HID * 4];   // transposed: sWt[n*4 + k]
    for (int i = threadIdx.x; i < 4 * HID; i += blockDim.x) {
        int k = i >> 7, n = i & 127;
        sWt[n * 4 + k] = W[i];
    }
    __syncthreads();

    const int wave  = threadIdx.x >> 5;
    const int lane  = threadIdx.x & 31;
    const int mbase = (blockIdx.x * 8 + wave) * 16;
    if (mbase >= nrows) return;

    const int mrow  = mbase + (lane & 15);
    const int khalf = (lane >> 4) * 2;
    const int ncol  = lane & 15;

    float2 t = *(const float2*)(X + (size_t)min(mrow, nrows - 1) * 4 + khalf);
    v2f aa; aa.x = t.x; aa.y = t.y;

    const int rbase = mbase + ((lane >> 4) << 3);
    for (int nt = 0; nt < 8; ++nt) {
        float2 tb = *(const float2*)(sWt + (size_t)(nt * 16 + ncol) * 4 + khalf);
        v2f b; b.x = tb.x; b.y = tb.y;
        v8f c = {};
        c = __builtin_amdgcn_wmma_f32_16x16x4_f32(
                false, aa, false, b, (short)0, c, false, false);
        #pragma unroll
        for (int r = 0; r < 8; ++r) {
            int row = rbase + r;
            if (row < nrows) Out[(size_t)row * HID + nt * 16 + ncol] = c[r];
        }
    }
}

// ---------------------------------------------------------------------------
// init: agg = h*dinv^2 + b      (self-loop + bias, before edge accumulation)
// ---------------------------------------------------------------------------
__global__ void init_kernel(
    float* __restrict__ agg, const float* __restrict__ h,
    const float* __restrict__ dinv, const float* __restrict__ b, int N)
{
    int idx = blockIdx.x * blockDim.x + threadIdx.x;   // one per (node, 4 feats)
    int node = idx >> 5;
    if (node >= N) return;
    int f = (idx & 31) * 4;
    float di = dinv[node];
    float sl = di * di;
    float4 vh = *(const float4*)(h + (size_t)node * HID + f);
    float4 vb = *(const float4*)(b + f);
    float4 o;
    o.x = vh.x * sl + vb.x;
    o.y = vh.y * sl + vb.y;
    o.z = vh.z * sl + vb.z;
    o.w = vh.w * sl + vb.w;
    *(float4*)(agg + (size_t)node * HID + f) = o;
}

// ---------------------------------------------------------------------------
// edge scatter: agg[dst] += h[src] * dinv[src]*dinv[dst]   (wave per edge)
// ---------------------------------------------------------------------------
__global__ __launch_bounds__(256) void scatter_kernel(
    const float* __restrict__ Hm, const int* __restrict__ src,
    const int* __restrict__ dst, const float* __restrict__ dinv,
    float* __restrict__ agg, int E)
{
    const int wave = threadIdx.x >> 5;
    const int lane = threadIdx.x & 31;
    const int e = blockIdx.x * 8 + wave;
    if (e >= E) return;
    const int s = src[e], d = dst[e];
    const float norm = dinv[s] * dinv[d];
    float4 v = *(const float4*)(Hm + (size_t)s * HID + lane * 4);
    float* o = agg + (size_t)d * HID + lane * 4;
    atomicAdd(o + 0, v.x * norm);
    atomicAdd(o + 1, v.y * norm);
    atomicAdd(o + 2, v.z * norm);
    atomicAdd(o + 3, v.w * norm);
}

// ---------------------------------------------------------------------------
// in-place ReLU
// ---------------------------------------------------------------------------
__global__ void relu_kernel(float4* __restrict__ p, int n4) {
    int i = blockIdx.x * blockDim.x + threadIdx.x;
    if (i >= n4) return;
    float4 v = p[i];
    v.x = fmaxf(v.x, 0.f);
    v.y = fmaxf(v.y, 0.f);
    v.z = fmaxf(v.z, 0.f);
    v.w = fmaxf(v.w, 0.f);
    p[i] = v;
}

// ---------------------------------------------------------------------------
// global mean pool (accumulate): gsum[batch[i]] += act[i]; gcnt[batch[i]] += 1
// ---------------------------------------------------------------------------
__global__ __launch_bounds__(256) void pool_kernel(
    const float* __restrict__ act, const int* __restrict__ batch,
    float* __restrict__ gsum, float* __restrict__ gcnt, int N)
{
    const int wave = threadIdx.x >> 5;
    const int lane = threadIdx.x & 31;
    const int node = blockIdx.x * 8 + wave;
    if (node >= N) return;
    const int g = batch[node];
    float4 v = *(const float4*)(act + (size_t)node * HID + lane * 4);
    float* o = gsum + (size_t)g * HID + lane * 4;
    atomicAdd(o + 0, v.x);
    atomicAdd(o + 1, v.y);
    atomicAdd(o + 2, v.z);
    atomicAdd(o + 3, v.w);
    if (lane == 0) atomicAdd(&gcnt[g], 1.0f);
}

// ---------------------------------------------------------------------------
// head: logits = (gsum/cnt) @ Wl + bl ; log_softmax    (one wave per graph)
// ---------------------------------------------------------------------------
__global__ void head_kernel(
    const float* __restrict__ gsum, const float* __restrict__ gcnt,
    const float* __restrict__ Wl, const float* __restrict__ bl,
    float* __restrict__ out)
{
    const int g = blockIdx.x;
    const int lane = threadIdx.x;
    const float inv = 1.0f / fmaxf(gcnt[g], 1.0f);
    float s0 = 0.f, s1 = 0.f;
    #pragma unroll
    for (int k = 0; k < 4; ++k) {
        int f = lane + 32 * k;
        float v = gsum[(size_t)g * HID + f] * inv;
        s0 += v * Wl[f * 2 + 0];
        s1 += v * Wl[f * 2 + 1];
    }
    #pragma unroll
    for (int off = 16; off > 0; off >>= 1) {
        s0 += __shfl_down(s0, off, 32);
        s1 += __shfl_down(s1, off, 32);
    }
    if (lane == 0) {
        float l0 = s0 + bl[0], l1 = s1 + bl[1];
        float m = fmaxf(l0, l1);
        float lse = m + logf(expf(l0 - m) + expf(l1 - m));
        out[g * 2 + 0] = l0 - lse;
        out[g * 2 + 1] = l1 - lse;
    }
}

// ---------------------------------------------------------------------------
// launch
// ---------------------------------------------------------------------------
static inline size_t alignf(size_t x) { return (x + 255) & ~(size_t)255; }

extern "C" void kernel_launch(void* const* d_in, const int* in_sizes, int n_in,
                              void* d_out, int out_size, void* d_ws, size_t ws_size,
                              hipStream_t stream) {
    const float* x    = (const float*)d_in[0];
    const int*   ei   = (const int*)  d_in[1];
    const int*   bat  = (const int*)  d_in[2];
    const float* W1   = (const float*)d_in[3];
    const float* b1   = (const float*)d_in[4];
    const float* W2   = (const float*)d_in[5];
    const float* b2   = (const float*)d_in[6];
    const float* W3   = (const float*)d_in[7];
    const float* b3   = (const float*)d_in[8];
    const float* Wl   = (const float*)d_in[9];
    const float* bl   = (const float*)d_in[10];
    float* out = (float*)d_out;

    const int N = in_sizes[0] / 4;       // 100000 nodes
    const int E = in_sizes[1] / 2;       // 1.6M edges
    const int* src = ei;
    const int* dst = ei + E;

    float* ws = (float*)d_ws;
    size_t off = 0;
    float* dinv = ws + off; off += alignf((size_t)N);
    float* hbuf = ws + off; off += alignf((size_t)N * HID);
    float* agg  = ws + off; off += alignf((size_t)N * HID);
    float* gsum = ws + off; off += alignf((size_t)NG * HID + NG);
    float* gcnt = gsum + NG * HID;

    const int TB = 256;
    const int mtiles   = (N + 15) / 16;
    const int gemmGrid = (mtiles + 7) / 8;
    const int edgeGrid = (E + 7) / 8;
    const int nodeGrid = (N + 7) / 8;
    const int nfGrid   = ((size_t)N * 32 + TB - 1) / TB;   // (node,4feat) grid
    const int n4       = N * HID / 4;
    const int n4Grid   = (n4 + TB - 1) / TB;

    // --- degrees / symmetric norm -----------------------------------------
    zero_kernel<<<(N / 4 + TB - 1) / TB, TB, 0, stream>>>((float4*)dinv, N / 4);
    deg_kernel<<<(E + TB - 1) / TB, TB, 0, stream>>>(dst, dinv, E);
    dinv_kernel<<<(N + TB - 1) / TB, TB, 0, stream>>>(dinv, N);

    // --- layer 1 ----------------------------------------------------------
    gemm4_kernel<<<gemmGrid, TB, 0, stream>>>(x, W1, hbuf, N);
    init_kernel<<<nfGrid, TB, 0, stream>>>(agg, hbuf, dinv, b1, N);
    scatter_kernel<<<edgeGrid, TB, 0, stream>>>(hbuf, src, dst, dinv, agg, E);
    relu_kernel<<<n4Grid, TB, 0, stream>>>((float4*)agg, n4);

    // --- layer 2 ----------------------------------------------------------
    gemm128_kernel<<<gemmGrid, TB, 0, stream>>>(agg, W2, hbuf, N);
    init_kernel<<<nfGrid, TB, 0, stream>>>(agg, hbuf, dinv, b2, N);
    scatter_kernel<<<edgeGrid, TB, 0, stream>>>(hbuf, src, dst, dinv, agg, E);
    relu_kernel<<<n4Grid, TB, 0, stream>>>((float4*)agg, n4);

    // --- layer 3 ----------------------------------------------------------
    gemm128_kernel<<<gemmGrid, TB, 0, stream>>>(agg, W3, hbuf, N);
    init_kernel<<<nfGrid, TB, 0, stream>>>(agg, hbuf, dinv, b3, N);
    scatter_kernel<<<edgeGrid, TB, 0, stream>>>(hbuf, src, dst, dinv, agg, E);
    relu_kernel<<<n4Grid, TB, 0, stream>>>((float4*)agg, n4);

    // --- pool + head ------------------------------------------------------
    int pz = (NG * HID + NG) / 4;
    zero_kernel<<<(pz + TB - 1) / TB, TB, 0, stream>>>((float4*)gsum, pz);
    pool_kernel<<<nodeGrid, TB, 0, stream>>>(agg, bat, gsum, gcnt, N);
    head_kernel<<<NG, 32, 0, stream>>>(gsum, gcnt, Wl, bl, out);
}